// BigBirdRegressor_MLP_42150809043590
// MI455X (gfx1250) — compile-verified
//
#include <hip/hip_runtime.h>
#include <hip/hip_bf16.h>
#include <stdint.h>

#define S_LEN   8192
#define NTOK    65536      // B*S
#define KDIM    24576      // S*h
#define HID1_D  1000
#define NTILES  63         // ceil(1000/16)
#define KSPLIT  32
#define KCHUNK  (KDIM / KSPLIT)   // 768

typedef float v2f __attribute__((ext_vector_type(2)));
typedef float v8f __attribute__((ext_vector_type(8)));

// ---------------------------------------------------------------------------
// Static BigBird random-block table: replicate np.random.RandomState(0)
// (legacy MT19937 + masked-rejection interval + Fisher-Yates permutation).
// Writes tab[124][8] = {0, 127, i-1, i, i+1, r0, r1, r2} for i in [2,125].
// ---------------------------------------------------------------------------
__global__ void k_randidx(int* __restrict__ tab) {
  __shared__ uint32_t mt[624];
  if (threadIdx.x != 0) return;
  uint32_t sd = 0u;
  for (int i = 0; i < 624; ++i) {
    mt[i] = sd;
    sd = 1812433253u * (sd ^ (sd >> 30)) + (uint32_t)(i + 1);
  }
  int pos = 624;
  int perm[123];
  for (int i = 2; i <= 125; ++i) {
    for (int t = 0; t < 123; ++t) perm[t] = t;
    for (int a = 122; a > 0; --a) {
      uint32_t mx = (uint32_t)a;
      uint32_t mask = mx;
      mask |= mask >> 1; mask |= mask >> 2; mask |= mask >> 4;
      mask |= mask >> 8; mask |= mask >> 16;
      uint32_t val;
      do {
        if (pos >= 624) {   // MT19937 twist (in-place, matches numpy order)
          for (int ii = 0; ii < 624; ++ii) {
            uint32_t y = (mt[ii] & 0x80000000u) | (mt[(ii + 1) % 624] & 0x7fffffffu);
            mt[ii] = mt[(ii + 397) % 624] ^ (y >> 1) ^ ((y & 1u) ? 0x9908b0dfu : 0u);
          }
          pos = 0;
        }
        uint32_t y = mt[pos++];
        y ^= y >> 11;
        y ^= (y << 7)  & 0x9d2c5680u;
        y ^= (y << 15) & 0xefc60000u;
        y ^= y >> 18;
        val = y & mask;
      } while (val > mx);
      int j = (int)val;
      int tmp = perm[a]; perm[a] = perm[j]; perm[j] = tmp;
    }
    int row = i - 2;
    tab[row * 8 + 0] = 0;
    tab[row * 8 + 1] = 127;
    tab[row * 8 + 2] = i - 1;
    tab[row * 8 + 3] = i;
    tab[row * 8 + 4] = i + 1;
    for (int t = 0; t < 3; ++t) {
      int c = perm[t];                        // index into sorted cand = [1..126]\{i-1,i,i+1}
      tab[row * 8 + 5 + t] = (c < i - 2) ? (c + 1) : (c + 4);
    }
  }
}

// ---------------------------------------------------------------------------
// LayerNorm over h=3.
// ---------------------------------------------------------------------------
__device__ __forceinline__ void ln3(float a0, float a1, float a2,
                                    const float* g, const float* b,
                                    float* o0, float* o1, float* o2) {
  float m = (a0 + a1 + a2) * (1.0f / 3.0f);
  float d0 = a0 - m, d1 = a1 - m, d2 = a2 - m;
  float var = (d0 * d0 + d1 * d1 + d2 * d2) * (1.0f / 3.0f);
  float inv = rsqrtf(var + 1e-12f);
  *o0 = d0 * inv * g[0] + b[0];
  *o1 = d1 * inv * g[1] + b[1];
  *o2 = d2 * inv * g[2] + b[2];
}

__global__ void k_embed(const float* __restrict__ emb, const float* __restrict__ pos,
                        const float* __restrict__ te, const float* __restrict__ g,
                        const float* __restrict__ bb, float* __restrict__ x) {
  int t = blockIdx.x * blockDim.x + threadIdx.x;
  if (t >= NTOK) return;
  int s = t & (S_LEN - 1);
  float a0 = emb[t * 3 + 0] + pos[s * 3 + 0] + te[0];
  float a1 = emb[t * 3 + 1] + pos[s * 3 + 1] + te[1];
  float a2 = emb[t * 3 + 2] + pos[s * 3 + 2] + te[2];
  float o0, o1, o2;
  ln3(a0, a1, a2, g, bb, &o0, &o1, &o2);
  x[t * 3 + 0] = o0; x[t * 3 + 1] = o1; x[t * 3 + 2] = o2;
}

// ---------------------------------------------------------------------------
// QKV projection (3x3 each); dh=1 so head j == channel j. Layout [b][head][s].
// ---------------------------------------------------------------------------
__global__ void k_qkv(const float* __restrict__ x,
                      const float* __restrict__ Wq, const float* __restrict__ bq,
                      const float* __restrict__ Wk, const float* __restrict__ bk,
                      const float* __restrict__ Wv, const float* __restrict__ bv,
                      float* __restrict__ q, float* __restrict__ k, float* __restrict__ v) {
  int t = blockIdx.x * blockDim.x + threadIdx.x;
  if (t >= NTOK) return;
  int b = t >> 13, s = t & (S_LEN - 1);
  float x0 = x[t * 3 + 0], x1 = x[t * 3 + 1], x2 = x[t * 3 + 2];
#pragma unroll
  for (int j = 0; j < 3; ++j) {
    int o = (b * 3 + j) * S_LEN + s;
    q[o] = x0 * Wq[j] + x1 * Wq[3 + j] + x2 * Wq[6 + j] + bq[j];
    k[o] = x0 * Wk[j] + x1 * Wk[3 + j] + x2 * Wk[6 + j] + bk[j];
    v[o] = x0 * Wv[j] + x1 * Wv[3 + j] + x2 * Wv[6 + j] + bv[j];
  }
}

// ---------------------------------------------------------------------------
// Online-softmax helpers (wave32).
// ---------------------------------------------------------------------------
__device__ __forceinline__ void osm_update(float sc, float vv, float& m, float& l, float& a) {
  float mn = fmaxf(m, sc);
  float e  = __expf(m - mn);
  float p  = __expf(sc - mn);
  l = l * e + p;
  a = a * e + p * vv;
  m = mn;
}

__device__ __forceinline__ void osm_reduce(float& m, float& l, float& a) {
#pragma unroll
  for (int off = 16; off; off >>= 1) {
    float m2 = __shfl_xor(m, off, 32);
    float l2 = __shfl_xor(l, off, 32);
    float a2 = __shfl_xor(a, off, 32);
    float mn = fmaxf(m, m2);
    float e1 = __expf(m - mn), e2 = __expf(m2 - mn);
    l = l * e1 + l2 * e2;
    a = a * e1 + a2 * e2;
    m = mn;
  }
}

// Global/boundary query blocks {0,1,126,127}: full attention over S. 1 wave/query.
__global__ void k_attn_global(const float* __restrict__ q, const float* __restrict__ k,
                              const float* __restrict__ v, float* __restrict__ ctx) {
  int w = (blockIdx.x * blockDim.x + threadIdx.x) >> 5;
  int lane = threadIdx.x & 31;
  if (w >= 8 * 3 * 256) return;
  int b = w / (3 * 256);
  int rem = w % (3 * 256);
  int hh = rem >> 8;
  int qi = rem & 255;
  int qb4 = qi >> 6;
  int blk = (qb4 < 2) ? qb4 : (124 + qb4);   // 0,1,126,127
  int sq = blk * 64 + (qi & 63);
  int base = (b * 3 + hh) * S_LEN;
  float qval = q[base + sq];                 // scale = 1/sqrt(dh) = 1
  float m = -3.0e38f, l = 0.f, acc = 0.f;
  for (int ks = lane; ks < S_LEN; ks += 32)
    osm_update(qval * k[base + ks], v[base + ks], m, l, acc);
  osm_reduce(m, l, acc);
  if (lane == 0) ctx[base + sq] = acc / l;
}

// Middle query blocks i in [2,125]: 8 gathered key blocks (512 keys). 1 wave/query.
__global__ void k_attn_mid(const float* __restrict__ q, const float* __restrict__ k,
                           const float* __restrict__ v, const int* __restrict__ tab,
                           float* __restrict__ ctx) {
  int w = (blockIdx.x * blockDim.x + threadIdx.x) >> 5;
  int lane = threadIdx.x & 31;
  if (w >= 8 * 3 * 124 * 64) return;
  int b = w / (3 * 124 * 64);
  int rem = w % (3 * 124 * 64);
  int hh = rem / (124 * 64);
  int rem2 = rem % (124 * 64);
  int row = rem2 >> 6;
  int qw = rem2 & 63;
  int sq = (row + 2) * 64 + qw;
  int base = (b * 3 + hh) * S_LEN;
  float qval = q[base + sq];
  float m = -3.0e38f, l = 0.f, acc = 0.f;
#pragma unroll 4
  for (int t = 0; t < 16; ++t) {
    int kk = lane + (t << 5);
    int blk = tab[row * 8 + (kk >> 6)];
    int sk = blk * 64 + (kk & 63);
    osm_update(qval * k[base + sk], v[base + sk], m, l, acc);
  }
  osm_reduce(m, l, acc);
  if (lane == 0) ctx[base + sq] = acc / l;
}

// ---------------------------------------------------------------------------
// Fused: attn-out proj + residual + LN1 + FFN(gelu_new) + residual + LN2.
// ---------------------------------------------------------------------------
__global__ void k_post(float* __restrict__ x, const float* __restrict__ ctx,
                       const float* __restrict__ Wo, const float* __restrict__ bo,
                       const float* __restrict__ g1, const float* __restrict__ b1,
                       const float* __restrict__ Wi, const float* __restrict__ bi,
                       const float* __restrict__ Wo2, const float* __restrict__ bo2,
                       const float* __restrict__ g2, const float* __restrict__ b2) {
  int t = blockIdx.x * blockDim.x + threadIdx.x;
  if (t >= NTOK) return;
  int b = t >> 13, s = t & (S_LEN - 1);
  float c0 = ctx[(b * 3 + 0) * S_LEN + s];
  float c1 = ctx[(b * 3 + 1) * S_LEN + s];
  float c2 = ctx[(b * 3 + 2) * S_LEN + s];
  float r0 = x[t * 3 + 0] + c0 * Wo[0] + c1 * Wo[3] + c2 * Wo[6] + bo[0];
  float r1 = x[t * 3 + 1] + c0 * Wo[1] + c1 * Wo[4] + c2 * Wo[7] + bo[1];
  float r2 = x[t * 3 + 2] + c0 * Wo[2] + c1 * Wo[5] + c2 * Wo[8] + bo[2];
  float x0, x1, x2;
  ln3(r0, r1, r2, g1, b1, &x0, &x1, &x2);
  float o0 = bo2[0], o1 = bo2[1], o2 = bo2[2];
#pragma unroll
  for (int j = 0; j < 12; ++j) {
    float raw = x0 * Wi[j] + x1 * Wi[12 + j] + x2 * Wi[24 + j] + bi[j];
    float f = 0.5f * raw * (1.0f + tanhf(0.7978845608028654f * (raw + 0.044715f * raw * raw * raw)));
    o0 += f * Wo2[j * 3 + 0];
    o1 += f * Wo2[j * 3 + 1];
    o2 += f * Wo2[j * 3 + 2];
  }
  ln3(x0 + o0, x1 + o1, x2 + o2, g2, b2, &x0, &x1, &x2);
  x[t * 3 + 0] = x0; x[t * 3 + 1] = x1; x[t * 3 + 2] = x2;
}

// ---------------------------------------------------------------------------
// MLP head GEMM: [16(pad of 8) x 24576] @ [24576 x 1008(pad of 1000)] in f32
// via v_wmma_f32_16x16x4_f32.
//  - K split KSPLIT ways for HBM concurrency; partials via global_atomic_add_f32.
//  - fc1_W streamed through LDS with double-buffered global_load_async_to_lds_b128
//    (2 x 512B coalesced bursts per 1KB chunk), synced with s_wait_asynccnt.
//  - No predicated loads: pad rows/cols use clamped in-bounds addresses; their
//    WMMA outputs land in D rows 8-15 / cols >=1000 which are never stored.
// ---------------------------------------------------------------------------
__global__ void k_head_gemm(const float* __restrict__ A, const float* __restrict__ W,
                            float* __restrict__ accb) {
  __shared__ float smem[2 * 256];   // 2 buffers x (16 K-rows x 16 cols)
  int tile  = blockIdx.x % NTILES;
  int split = blockIdx.x / NTILES;
  int lane  = threadIdx.x & 31;
  int mn    = lane & 15;               // m for A-frag, n for B-frag/C-tile
  int khalf = (lane >> 4) << 1;        // 0 or 2
  int n     = tile * 16 + mn;
  const float* arow = A + (mn & 7) * KDIM;   // rows 8-15 alias 0-7 (outputs unused)

  // Staging geometry: lane -> (row 0..7, 4-float col group); 2 issues cover 16 rows.
  int srow  = lane >> 2;               // 0..7
  int scol  = (lane & 3) << 2;         // 0,4,8,12
  int colb  = tile * 16 + scol;
  if (colb > HID1_D - 4) colb = HID1_D - 4;    // clamp in-bounds (dup cols unused)
  uint32_t ldsbase = (uint32_t)(uintptr_t)(&smem[0]);
  uint32_t lds0 = ldsbase + (uint32_t)(((srow)     * 16 + scol) * 4);
  uint32_t lds1 = ldsbase + (uint32_t)(((srow + 8) * 16 + scol) * 4);

  int k0 = split * KCHUNK;
  int kend = k0 + KCHUNK;

  {ature:; }  // (no-op label removed by preprocessor? -- keep code simple below)

  // initial stage into buffer 0
  {
    uint64_t g0 = (uint64_t)(uintptr_t)(W + (size_t)(k0 + srow) * HID1_D + colb);
    uint64_t g1 = g0 + (uint64_t)(8 * HID1_D) * 4u;
    asm volatile("global_load_async_to_lds_b128 %0, %1, off"
                 :: "v"(lds0), "v"(g0) : "memory");
    asm volatile("global_load_async_to_lds_b128 %0, %1, off"
                 :: "v"(lds1), "v"(g1) : "memory");
  }

  v8f c = {};
  int buf = 0;
  for (int kb = k0; kb < kend; kb += 16) {
    if (kb + 128 < KDIM)
      __builtin_prefetch(W + (size_t)(kb + 128) * HID1_D + colb, 0, 1);
    if (kb + 16 < kend) {
      uint32_t boff = (uint32_t)((buf ^ 1) * 1024);
      uint64_t g0 = (uint64_t)(uintptr_t)(W + (size_t)(kb + 16 + srow) * HID1_D + colb);
      uint64_t g1 = g0 + (uint64_t)(8 * HID1_D) * 4u;
      asm volatile("global_load_async_to_lds_b128 %0, %1, off"
                   :: "v"(lds0 + boff), "v"(g0) : "memory");
      asm volatile("global_load_async_to_lds_b128 %0, %1, off"
                   :: "v"(lds1 + boff), "v"(g1) : "memory");
      asm volatile("s_wait_asynccnt 0x2" ::: "memory");  // current buffer complete
    } else {
      asm volatile("s_wait_asynccnt 0x0" ::: "memory");
    }
    const float* sb = &smem[buf * 256];
#pragma unroll
    for (int u = 0; u < 4; ++u) {
      int kl = u * 4 + khalf;
      v2f a = *(const v2f*)(arow + kb + kl);      // global_load_b64, unconditional
      v2f bf;
      bf.x = sb[kl * 16 + mn];                    // ds_load, conflict-free
      bf.y = sb[kl * 16 + 16 + mn];
      c = __builtin_amdgcn_wmma_f32_16x16x4_f32(false, a, false, bf,
                                                (short)0, c, false, false);
    }
    buf ^= 1;
  }

  if (lane < 16 && n < HID1_D) {
#pragma unroll
    for (int r = 0; r < 8; ++r)
      atomicAdd(&accb[r * HID1_D + n], c[r]);
  }
}

__global__ void k_zero(float* __restrict__ p, int nelem) {
  int t = blockIdx.x * blockDim.x + threadIdx.x;
  if (t < nelem) p[t] = 0.0f;
}

__global__ void k_head_bn(const float* __restrict__ accb, const float* __restrict__ fb,
                          const float* __restrict__ bng, const float* __restrict__ bnb,
                          const float* __restrict__ bnm, const float* __restrict__ bnv,
                          float* __restrict__ y1) {
  int t = blockIdx.x * blockDim.x + threadIdx.x;
  if (t >= 8 * HID1_D) return;
  int nn = t % HID1_D;
  float y = accb[t] + fb[nn];
  y = (y - bnm[nn]) * rsqrtf(bnv[nn] + 1e-5f) * bng[nn] + bnb[nn];
  y1[t] = fmaxf(y, 0.0f);
}

__global__ void k_out(const float* __restrict__ y1, const float* __restrict__ w2,
                      const float* __restrict__ b2, float* __restrict__ out) {
  int b = threadIdx.x >> 5, lane = threadIdx.x & 31;
  float s = 0.0f;
  for (int nn = lane; nn < HID1_D; nn += 32) s += y1[b * HID1_D + nn] * w2[nn];
#pragma unroll
  for (int off = 16; off; off >>= 1) s += __shfl_xor(s, off, 32);
  if (lane == 0) out[b] = s + b2[0];
}

// ---------------------------------------------------------------------------
extern "C" void kernel_launch(void* const* d_in, const int* in_sizes, int n_in,
                              void* d_out, int out_size, void* d_ws, size_t ws_size,
                              hipStream_t stream) {
  (void)in_sizes; (void)n_in; (void)out_size; (void)ws_size;
  const float* inputs_embeds = (const float*)d_in[0];
  const float* pos_emb = (const float*)d_in[1];
  const float* type_emb = (const float*)d_in[2];
  const float* ln_e_g = (const float*)d_in[3];
  const float* ln_e_b = (const float*)d_in[4];
  const float* Wq = (const float*)d_in[5];
  const float* bq = (const float*)d_in[6];
  const float* Wk = (const float*)d_in[7];
  const float* bk = (const float*)d_in[8];
  const float* Wv = (const float*)d_in[9];
  const float* bv = (const float*)d_in[10];
  const float* Wo = (const float*)d_in[11];
  const float* bo = (const float*)d_in[12];
  const float* ln1_g = (const float*)d_in[13];
  const float* ln1_b = (const float*)d_in[14];
  const float* Wi = (const float*)d_in[15];
  const float* bi = (const float*)d_in[16];
  const float* Wo2 = (const float*)d_in[17];
  const float* bo2 = (const float*)d_in[18];
  const float* ln2_g = (const float*)d_in[19];
  const float* ln2_b = (const float*)d_in[20];
  const float* fc1_W = (const float*)d_in[21];
  const float* fc1_b = (const float*)d_in[22];
  const float* bn_g = (const float*)d_in[23];
  const float* bn_b = (const float*)d_in[24];
  const float* bn_mean = (const float*)d_in[25];
  const float* bn_var = (const float*)d_in[26];
  const float* fc2_W = (const float*)d_in[27];
  const float* fc2_b = (const float*)d_in[28];

  float* ws  = (float*)d_ws;
  float* x    = ws;                 // 196608
  float* q    = ws + 196608;        // 196608
  float* k    = ws + 393216;        // 196608
  float* v    = ws + 589824;        // 196608
  float* ctx  = ws + 786432;        // 196608
  float* accb = ws + 983040;        // 8192
  float* y1   = ws + 991232;        // 8192
  int* idxtab = (int*)(ws + 999424);// 992 ints

  k_randidx<<<1, 32, 0, stream>>>(idxtab);
  k_embed<<<256, 256, 0, stream>>>(inputs_embeds, pos_emb, type_emb, ln_e_g, ln_e_b, x);
  for (int l = 0; l < 2; ++l) {
    k_qkv<<<256, 256, 0, stream>>>(x, Wq + 9 * l, bq + 3 * l, Wk + 9 * l, bk + 3 * l,
                                   Wv + 9 * l, bv + 3 * l, q, k, v);
    k_attn_global<<<768, 256, 0, stream>>>(q, k, v, ctx);
    k_attn_mid<<<23808, 256, 0, stream>>>(q, k, v, idxtab, ctx);
    k_post<<<256, 256, 0, stream>>>(x, ctx, Wo + 9 * l, bo + 3 * l, ln1_g + 3 * l,
                                    ln1_b + 3 * l, Wi + 36 * l, bi + 12 * l,
                                    Wo2 + 36 * l, bo2 + 3 * l, ln2_g + 3 * l, ln2_b + 3 * l);
  }
  k_zero<<<32, 256, 0, stream>>>(accb, 8192);
  k_head_gemm<<<NTILES * KSPLIT, 32, 0, stream>>>(x, fc1_W, accb);
  k_head_bn<<<32, 256, 0, stream>>>(accb, fc1_b, bn_g, bn_b, bn_mean, bn_var, y1);
  k_out<<<1, 256, 0, stream>>>(y1, fc2_W, fc2_b, (float*)d_out);
}